// Net_35905926594573
// MI455X (gfx1250) — compile-verified
//
#include <hip/hip_runtime.h>

// GCN dims (fixed by the reference)
#define F_IN  128
#define F_HID 64
#define F_OUT 128

typedef __attribute__((ext_vector_type(2))) float v2f;
typedef __attribute__((ext_vector_type(8))) float v8f;

// ---------------------------------------------------------------------------
// Degree / normalization kernels
// ---------------------------------------------------------------------------
__global__ void deg_init_kernel(float* __restrict__ deg, int n) {
    int i = blockIdx.x * blockDim.x + threadIdx.x;
    if (i < n) deg[i] = 1.0f;  // self-loop contributes 1 to in-degree
}

__global__ void deg_accum_kernel(const int* __restrict__ dst, float* __restrict__ deg, int e) {
    int i = blockIdx.x * blockDim.x + threadIdx.x;
    if (i < e) unsafeAtomicAdd(&deg[dst[i]], 1.0f);
}

__global__ void dinv_kernel(float* __restrict__ deg, int n) {
    int i = blockIdx.x * blockDim.x + threadIdx.x;
    if (i < n) deg[i] = rsqrtf(deg[i]);  // deg >= 1 always (self-loop)
}

__global__ void edge_norm_kernel(const int* __restrict__ src, const int* __restrict__ dst,
                                 const float* __restrict__ dinv, float* __restrict__ normE,
                                 int e) {
    int i = blockIdx.x * blockDim.x + threadIdx.x;
    if (i < e) normE[i] = dinv[src[i]] * dinv[dst[i]];
}

// ---------------------------------------------------------------------------
// f32 WMMA GEMM:  C[M,N] = op(A)[M,K] * B[K,N],  op = identity or ReLU.
// One wave computes one 16x16 tile of C using V_WMMA_F32_16X16X4_F32.
// blockDim.x = (N/16)*32 : wave w owns column tile w; blockIdx.x = row tile.
//
// A-fragment (16x4 f32, ISA 7.12.2): lane L -> m = L&15, K pair = 2*(L>>4)
// B-fragment (4x16 f32): lane L -> n = L&15, same K pair
// C/D (16x16 f32, 8 VGPRs): VGPR r -> row r (lanes 0-15) / row r+8 (lanes 16-31)
// ---------------------------------------------------------------------------
template <int K, int N, bool RELU>
__global__ void gemm_wmma_f32(const float* __restrict__ A, const float* __restrict__ B,
                              float* __restrict__ C, int M) {
    const int lane  = threadIdx.x & 31;
    const int wave  = threadIdx.x >> 5;
    const int m     = lane & 15;   // row-in-tile (A) / col-in-tile (B, C)
    const int kh    = lane >> 4;   // which K pair: 0 -> k{0,1}, 1 -> k{2,3}
    const int kbase = kh * 2;
    const int row0  = blockIdx.x * 16;
    const int col0  = wave * 16;
    if (row0 >= M) return;  // block-uniform: whole wave exits together

    int ra = row0 + m;
    if (ra >= M) ra = M - 1;  // clamp (safe dup read; store is guarded)
    const float* __restrict__ arow = A + (size_t)ra * K;

    v8f acc = {};
    for (int kk = 0; kk < K; kk += 4) {
        v2f a;
        if (RELU) {
            a.x = fmaxf(arow[kk + kbase], 0.0f);
            a.y = fmaxf(arow[kk + kbase + 1], 0.0f);
        } else {
            a = *(const v2f*)(arow + kk + kbase);  // 8B-aligned: kk%4==0, kbase even
        }
        v2f b;
        b.x = B[(size_t)(kk + kbase) * N + col0 + m];
        b.y = B[(size_t)(kk + kbase + 1) * N + col0 + m];
        acc = __builtin_amdgcn_wmma_f32_16x16x4_f32(
            /*neg_a=*/false, a, /*neg_b=*/false, b,
            /*c_mod=*/(short)0, acc, /*reuse_a=*/false, /*reuse_b=*/false);
    }

#pragma unroll
    for (int r = 0; r < 8; ++r) {
        int row = row0 + r + kh * 8;
        if (row < M) C[(size_t)row * N + col0 + m] = acc[r];
    }
}

// ---------------------------------------------------------------------------
// Aggregation: out = D^-1/2 (A+I) D^-1/2 h + bias
//   init : out[i][f] = h[i][f] * dinv[i]^2 + bias[f]       (self-loop + bias)
//   edge : out[dst][f] += h[src][f] * norm[e]               (atomic scatter)
// F is a power of two (64 or 128); SHIFT = log2(F). Lane-per-(edge,feature):
// consecutive lanes cover consecutive features -> coalesced gather + atomics.
// ---------------------------------------------------------------------------
template <int F, int SHIFT>
__global__ void agg_init_kernel(const float* __restrict__ h, const float* __restrict__ dinv,
                                const float* __restrict__ bias, float* __restrict__ out,
                                unsigned total) {
    unsigned idx = blockIdx.x * blockDim.x + threadIdx.x;
    if (idx < total) {
        unsigned i = idx >> SHIFT;
        unsigned f = idx & (F - 1);
        float di = dinv[i];
        out[idx] = h[idx] * di * di + bias[f];
    }
}

template <int F, int SHIFT>
__global__ void agg_edge_kernel(const int* __restrict__ src, const int* __restrict__ dst,
                                const float* __restrict__ normE, const float* __restrict__ h,
                                float* __restrict__ out, unsigned e) {
    unsigned idx = blockIdx.x * blockDim.x + threadIdx.x;
    unsigned ei  = idx >> SHIFT;
    if (ei < e) {
        unsigned f = idx & (F - 1);
        float v = h[(size_t)src[ei] * F + f] * normE[ei];
        unsafeAtomicAdd(&out[(size_t)dst[ei] * F + f], v);
    }
}

// ---------------------------------------------------------------------------
// Launch
// ---------------------------------------------------------------------------
extern "C" void kernel_launch(void* const* d_in, const int* in_sizes, int n_in,
                              void* d_out, int out_size, void* d_ws, size_t ws_size,
                              hipStream_t stream) {
    const float* x  = (const float*)d_in[0];
    const int*   ei = (const int*)d_in[1];
    const float* W1 = (const float*)d_in[2];  // [128, 64] row-major
    const float* b1 = (const float*)d_in[3];
    const float* W2 = (const float*)d_in[4];  // [64, 128] row-major
    const float* b2 = (const float*)d_in[5];
    float* out = (float*)d_out;               // [n, 128]

    const int n = in_sizes[0] / F_IN;
    const int e = in_sizes[1] / 2;
    const int* src = ei;       // edge_index[0]
    const int* dst = ei + e;   // edge_index[1]

    // Workspace partition (all f32), 256B aligned.
    auto aup = [](size_t v) { return (v + 255) & ~(size_t)255; };
    char* p = (char*)d_ws;
    float* dinv  = (float*)p; p += aup((size_t)n * 4);          //  0.4 MB
    float* normE = (float*)p; p += aup((size_t)e * 4);          // 12.8 MB
    float* h1    = (float*)p; p += aup((size_t)n * F_HID * 4);  // 25.6 MB
    float* agg1  = (float*)p; p += aup((size_t)n * F_HID * 4);  // 25.6 MB
    float* h2    = (float*)p; p += aup((size_t)n * F_OUT * 4);  // 51.2 MB

    const int T = 256;

    // Normalization (computed once, shared by both layers)
    deg_init_kernel <<<(n + T - 1) / T, T, 0, stream>>>(dinv, n);
    deg_accum_kernel<<<(e + T - 1) / T, T, 0, stream>>>(dst, dinv, e);
    dinv_kernel     <<<(n + T - 1) / T, T, 0, stream>>>(dinv, n);
    edge_norm_kernel<<<(e + T - 1) / T, T, 0, stream>>>(src, dst, dinv, normE, e);

    // Layer 1: h1 = x @ W1 ; agg1 = norm-propagate(h1) + b1
    gemm_wmma_f32<F_IN, F_HID, false>
        <<<(n + 15) / 16, (F_HID / 16) * 32, 0, stream>>>(x, W1, h1, n);
    {
        unsigned total = (unsigned)n * F_HID;
        agg_init_kernel<F_HID, 6><<<(total + T - 1) / T, T, 0, stream>>>(h1, dinv, b1, agg1, total);
        unsigned etot = (unsigned)e * F_HID;
        agg_edge_kernel<F_HID, 6><<<(etot + T - 1) / T, T, 0, stream>>>(src, dst, normE, h1, agg1, (unsigned)e);
    }

    // Layer 2: h2 = relu(agg1) @ W2 (ReLU fused into A load); out = propagate(h2) + b2
    gemm_wmma_f32<F_HID, F_OUT, true>
        <<<(n + 15) / 16, (F_OUT / 16) * 32, 0, stream>>>(agg1, W2, h2, n);
    {
        unsigned total = (unsigned)n * F_OUT;
        agg_init_kernel<F_OUT, 7><<<(total + T - 1) / T, T, 0, stream>>>(h2, dinv, b2, out, total);
        unsigned etot = (unsigned)e * F_OUT;
        agg_edge_kernel<F_OUT, 7><<<(etot + T - 1) / T, T, 0, stream>>>(src, dst, normE, h2, out, (unsigned)e);
    }
}